// PITSiSNRLoss_65670049956214
// MI455X (gfx1250) — compile-verified
//
#include <hip/hip_runtime.h>
#include <hip/hip_bf16.h>
#include <math.h>

typedef float v2f __attribute__((ext_vector_type(2)));
typedef float v4f __attribute__((ext_vector_type(4)));
typedef float v8f __attribute__((ext_vector_type(8)));

// Problem constants (fixed by the reference): output_audios (2,64,256000), s1/s2 (64,256000)
#define BATCH 64
#define SIGLEN 256000
#define NGROUP 16     // 64 batches / 4 per group -> 16 row-groups of 16 signals
#define NCHUNK 125    // L-chunks per group; 125 chunks * 8 waves * 256 k = 256000
#define WAVE_K 256    // contiguous k-elements per wave
#define WAVE_IT (WAVE_K / 8)

// ---------------------------------------------------------------------------
// Kernel 0: zero the 16 x 16 x 16 Gram accumulator table (16*256 floats) in ws
// ---------------------------------------------------------------------------
__global__ void PIT_zero_kernel(float* __restrict__ gram) {
    gram[blockIdx.x * 256 + threadIdx.x] = 0.0f;
}

// ---------------------------------------------------------------------------
// Kernel 1: batched Gram via V_WMMA_F32_16X16X4_F32.
// Rows of X (per group g): row r -> batch b = 4g + (r>>2), signal = r&3
//   sig 0: output_audios[0][b]   sig 1: output_audios[1][b]
//   sig 2: s1[b]                 sig 3: s2[b]
// Each wave: 32 iterations, each = one b128 NT load/lane (16 rows x 8 k),
// four half-wave shfl_xor exchanges to build the 16x4 A fragments, two WMMAs.
// A == B (Gram symmetry + matching K-striping of the A and B f32 layouts).
// ---------------------------------------------------------------------------
__global__ __launch_bounds__(256) void PIT_gram_kernel(
    const float* __restrict__ out_audios,
    const float* __restrict__ s1,
    const float* __restrict__ s2,
    float* __restrict__ gram)
{
    const int g    = blockIdx.x;            // batch-group 0..15
    const int wave = threadIdx.x >> 5;      // 0..7
    const int lane = threadIdx.x & 31;
    const int hf   = lane >> 4;             // lane half (0: lanes 0-15, 1: 16-31)
    const int row  = lane & 15;             // matrix row this lane feeds

    const int wv = blockIdx.y * 8 + wave;   // 0..999 wave index within group
    const int k0 = wv * WAVE_K;

    const int b   = g * 4 + (row >> 2);
    const int sig = row & 3;
    const float* base =
        (sig == 0) ? (out_audios + (size_t)b * SIGLEN) :
        (sig == 1) ? (out_audios + (size_t)(BATCH + b) * SIGLEN) :
        (sig == 2) ? (s1 + (size_t)b * SIGLEN) :
                     (s2 + (size_t)b * SIGLEN);
    const float* p = base + k0 + 4 * hf;    // this lane streams p[it*8 .. it*8+3]

    v8f acc = {};                           // 16x16 f32 Gram accumulator (C/D)

    #pragma unroll 4
    for (int it = 0; it < WAVE_IT; ++it) {
        const v4f* vp = (const v4f*)(p + it * 8);
        v4f v = __builtin_nontemporal_load(vp);          // streamed once: NT hint
        // prefetch ~32 iterations (256 floats) ahead; speculative, drop-safe
        __builtin_prefetch((const void*)(p + it * 8 + 256), 0, 0);

        // half-wave exchange (lane <-> lane^16) to build A fragments
        float sw0 = __shfl_xor(v.x, 16, 32);
        float sw1 = __shfl_xor(v.y, 16, 32);
        float sw2 = __shfl_xor(v.z, 16, 32);
        float sw3 = __shfl_xor(v.w, 16, 32);

        // Fragment for K = kbase..kbase+3:
        //   VGPR0: lanes<16 -> X[r][k+0], lanes>=16 -> X[r][k+2]
        //   VGPR1: lanes<16 -> X[r][k+1], lanes>=16 -> X[r][k+3]
        v2f f0, f1;
        f0.x = hf ? sw2 : v.x;
        f0.y = hf ? sw3 : v.y;
        // Fragment for K = kbase+4..kbase+7:
        f1.x = hf ? v.z : sw0;
        f1.y = hf ? v.w : sw1;

        // D = A * A^T + C  (same registers as A and B: symmetric Gram)
        acc = __builtin_amdgcn_wmma_f32_16x16x4_f32(
            false, f0, false, f0, (short)0, acc, false, false);
        acc = __builtin_amdgcn_wmma_f32_16x16x4_f32(
            false, f1, false, f1, (short)0, acc, false, false);
    }

    // --- combine 8 waves of this workgroup in LDS (ds_add_f32) ---
    __shared__ float lacc[256];
    lacc[threadIdx.x] = 0.0f;
    __syncthreads();
    // C/D layout: VGPR v holds (M = v + 8*hf, N = row)
    #pragma unroll
    for (int v = 0; v < 8; ++v) {
        int M = v + 8 * hf;
        atomicAdd(&lacc[M * 16 + row], acc[v]);
    }
    __syncthreads();

    // --- one global f32 atomic per needed entry (block-diagonal 4x4 blocks) ---
    int t = threadIdx.x;
    int M = t >> 4, N = t & 15;
    if ((M >> 2) == (N >> 2)) {
        atomicAdd(&gram[g * 256 + t], lacc[t]);
    }
}

// ---------------------------------------------------------------------------
// Kernel 2: per-batch SI-SDR + PIT max + mean; single 64-thread block.
// ---------------------------------------------------------------------------
__global__ __launch_bounds__(64) void PIT_finalize_kernel(
    const float* __restrict__ gram, float* __restrict__ out)
{
    const float EPS = 1.1920928955078125e-7f;   // float32 machine epsilon
    int t = threadIdx.x;                        // batch index 0..63
    int g = t >> 2, j = t & 3;
    const float* G = gram + g * 256;
    int r = 4 * j;

    auto at = [&](int m, int n) { return G[(r + m) * 16 + (r + n)]; };
    float E0 = at(0, 0), E1 = at(1, 1), E2 = at(2, 2), E3 = at(3, 3);
    float d02 = at(0, 2), d13 = at(1, 3), d12 = at(1, 2), d03 = at(0, 3);

    auto sdr = [&](float dot, float tt, float pp) {
        float alpha = (dot + EPS) / (tt + EPS);
        float ts    = alpha * alpha * tt;                 // ||alpha*t||^2
        float noise = pp - 2.0f * alpha * dot + ts;       // ||p - alpha*t||^2
        return 10.0f * __log10f((ts + EPS) / (noise + EPS));
    };

    float score1 = 0.5f * (sdr(d02, E2, E0) + sdr(d13, E3, E1)); // (p0,s1),(p1,s2)
    float score2 = 0.5f * (sdr(d12, E2, E1) + sdr(d03, E3, E0)); // (p1,s1),(p0,s2)
    float per    = fmaxf(score1, score2);

    __shared__ float red[64];
    red[t] = per;
    __syncthreads();
    #pragma unroll
    for (int s = 32; s > 0; s >>= 1) {
        if (t < s) red[t] += red[t + s];
        __syncthreads();
    }
    if (t == 0) out[0] = -red[0] / (float)BATCH;
}

// ---------------------------------------------------------------------------
extern "C" void kernel_launch(void* const* d_in, const int* in_sizes, int n_in,
                              void* d_out, int out_size, void* d_ws, size_t ws_size,
                              hipStream_t stream) {
    (void)in_sizes; (void)n_in; (void)out_size; (void)ws_size;
    const float* out_audios = (const float*)d_in[0];   // (2, 64, 256000) f32
    const float* s1         = (const float*)d_in[1];   // (64, 256000) f32
    const float* s2         = (const float*)d_in[2];   // (64, 256000) f32
    float* gram = (float*)d_ws;                        // 16 groups * 256 floats = 16 KB
    float* out  = (float*)d_out;                       // scalar loss

    PIT_zero_kernel<<<NGROUP, 256, 0, stream>>>(gram);
    dim3 grid(NGROUP, NCHUNK);
    PIT_gram_kernel<<<grid, 256, 0, stream>>>(out_audios, s1, s2, gram);
    PIT_finalize_kernel<<<1, 64, 0, stream>>>(gram, out);
}